// FeatureSimilarityLoss_25323127177734
// MI455X (gfx1250) — compile-verified
//
#include <hip/hip_runtime.h>

typedef __attribute__((ext_vector_type(2))) float v2f;
typedef __attribute__((ext_vector_type(8))) float v8f;

#define D_FEAT 128

// ---------------------------------------------------------------------------
// Kernel 1: zero the workspace accumulators (sum_wf, sum_w, sum_wn, acc, cnt)
// ---------------------------------------------------------------------------
__global__ void fsl_zero_kernel(float* __restrict__ ws, long long n,
                                double* __restrict__ acc, unsigned* __restrict__ cnt) {
  long long i = (long long)blockIdx.x * blockDim.x + threadIdx.x;
  long long stride = (long long)gridDim.x * blockDim.x;
  for (; i < n; i += stride) ws[i] = 0.0f;
  if (blockIdx.x == 0 && threadIdx.x == 0) { *acc = 0.0; *cnt = 0u; }
}

// ---------------------------------------------------------------------------
// Kernel 2: one wave (32 lanes) per edge.
//   lane handles 4 consecutive feature dims (float4 gather, 4 f32 atomics).
//   Accumulates: sum_wf[s,:] += w*x, sum_w[s] += w, sum_wn[s] += w*||x||^2.
// All accumulators are L2-resident (25.6 MB << 192 MB L2).
// ---------------------------------------------------------------------------
__global__ void fsl_pass1_kernel(const float* __restrict__ ew,
                                 const float* __restrict__ feat,
                                 const int* __restrict__ eidx,
                                 float* __restrict__ sum_wf,
                                 float* __restrict__ sum_w,
                                 float* __restrict__ sum_wn,
                                 int E) {
  int wave = (blockIdx.x * blockDim.x + threadIdx.x) >> 5;
  int lane = threadIdx.x & 31;
  if (wave >= E) return;  // wave-uniform exit

  float w = ew[wave];
  int s = eidx[wave];       // edge_index[0][e]
  int a = eidx[E + wave];   // edge_index[1][e]

  const float4 x = *((const float4*)(feat + (size_t)a * D_FEAT) + lane);

  float* dst = sum_wf + (size_t)s * D_FEAT + lane * 4;
  atomicAdd(dst + 0, w * x.x);
  atomicAdd(dst + 1, w * x.y);
  atomicAdd(dst + 2, w * x.z);
  atomicAdd(dst + 3, w * x.w);

  float wn = w * (x.x * x.x + x.y * x.y + x.z * x.z + x.w * x.w);
#pragma unroll
  for (int off = 16; off > 0; off >>= 1) wn += __shfl_down(wn, off, 32);

  if (lane == 0) {
    atomicAdd(sum_w + s, w);
    atomicAdd(sum_wn + s, wn);
  }
}

// ---------------------------------------------------------------------------
// Kernel 3: finalize. Each wave computes ||sum_wf[s]||^2 for 16 sources using
// V_WMMA_F32_16X16X4_F32: D = F * F^T accumulated over 32 chunks of K=4;
// the 16 self-dots are the diagonal of D. Then
//   var_s = sum_wn_s - dot_s*inv*(2 - sum_w_s*inv),  inv = 1/(sum_w_s+1e-8)
// reduced into a double accumulator; valid sources counted.
// ---------------------------------------------------------------------------
__global__ void fsl_finalize_kernel(const float* __restrict__ sum_wf,
                                    const float* __restrict__ sum_w,
                                    const float* __restrict__ sum_wn,
                                    double* __restrict__ acc,
                                    unsigned* __restrict__ cnt,
                                    int NS) {
  int wave = (blockIdx.x * blockDim.x + threadIdx.x) >> 5;
  int lane = threadIdx.x & 31;
  int base = wave * 16;
  if (base >= NS) return;  // wave-uniform exit; EXEC all-ones for WMMA below

  // A-matrix 16x4 f32 layout (ISA 7.12.2): lanes 0-15 / 16-31 both hold rows
  // M=0..15; VGPR0 = K{0|2}, VGPR1 = K{1|3}. B (4x16) mirrors with N in the
  // lane position, so for the symmetric self-product A and B register data
  // are identical.
  int row = base + (lane & 15);
  if (row >= NS) row = NS - 1;              // tail clamp (masked out below)
  int ksub = (lane >> 4) * 2;               // 0 for lanes 0-15, 2 for 16-31
  const float* fr = sum_wf + (size_t)row * D_FEAT + ksub;

  v8f c = {0.f, 0.f, 0.f, 0.f, 0.f, 0.f, 0.f, 0.f};
#pragma unroll
  for (int kc = 0; kc < D_FEAT / 4; ++kc) {
    v2f ab = *(const v2f*)(fr + kc * 4);
    c = __builtin_amdgcn_wmma_f32_16x16x4_f32(
        /*neg_a=*/false, ab, /*neg_b=*/false, ab,
        /*c_mod=*/(short)0, c, /*reuse_a=*/false, /*reuse_b=*/false);
  }

  // Diagonal extraction per 16x16 f32 C/D layout:
  //   m<8 : VGPR m,  lane m      (N = lane)
  //   m>=8: VGPR m-8, lane m+16  (N = lane-16)
  union { v8f v; float f[8]; } cu; cu.v = c;
  float diag = 0.0f; int m = -1;
  if (lane < 8)        { m = lane;      diag = cu.f[lane]; }
  else if (lane >= 24) { m = lane - 16; diag = cu.f[lane - 24]; }

  float var = 0.0f; unsigned ok = 0u;
  if (m >= 0) {
    int s = base + m;
    if (s < NS) {
      float sw = sum_w[s];
      if (sw > 0.0f) {
        float inv = 1.0f / (sw + 1e-8f);
        var = sum_wn[s] - diag * inv * (2.0f - sw * inv);
        ok = 1u;
      }
    }
  }
#pragma unroll
  for (int off = 16; off > 0; off >>= 1) {
    var += __shfl_down(var, off, 32);
    ok  += __shfl_down(ok,  off, 32);
  }
  if (lane == 0) {
    atomicAdd(acc, (double)var);
    atomicAdd(cnt, ok);
  }
}

// ---------------------------------------------------------------------------
// Kernel 4: scalar writeback: loss = acc / n_valid (0 if none valid)
// ---------------------------------------------------------------------------
__global__ void fsl_writeback_kernel(const double* __restrict__ acc,
                                     const unsigned* __restrict__ cnt,
                                     float* __restrict__ out) {
  if (blockIdx.x == 0 && threadIdx.x == 0) {
    unsigned c = *cnt;
    out[0] = (c > 0u) ? (float)(*acc / (double)c) : 0.0f;
  }
}

extern "C" void kernel_launch(void* const* d_in, const int* in_sizes, int n_in,
                              void* d_out, int out_size, void* d_ws, size_t ws_size,
                              hipStream_t stream) {
  const float* ew   = (const float*)d_in[0];  // edge_weights [E]
  const float* feat = (const float*)d_in[1];  // agent_features [N_AGENTS, 128]
  const int*   eidx = (const int*)d_in[2];    // edge_index [2, E] (int32)

  const int E  = in_sizes[0];
  const int NS = in_sizes[1] / D_FEAT;        // == num_s for this problem (50000)

  // Workspace layout: sum_wf[NS*128] | sum_w[NS] | sum_wn[NS] | acc(d) | cnt(u32)
  float* sum_wf = (float*)d_ws;
  float* sum_w  = sum_wf + (size_t)NS * D_FEAT;
  float* sum_wn = sum_w + NS;
  size_t foff = (size_t)NS * D_FEAT + 2 * (size_t)NS;
  foff = (foff + 1) & ~(size_t)1;             // 8-byte align for double
  double*   acc = (double*)(sum_wf + foff);
  unsigned* cnt = (unsigned*)(acc + 1);

  long long zn = (long long)NS * D_FEAT + 2LL * NS;
  fsl_zero_kernel<<<2048, 256, 0, stream>>>(sum_wf, zn, acc, cnt);

  int blocks1 = (E + 7) / 8;                  // 8 waves of 32 per 256-thread block
  fsl_pass1_kernel<<<blocks1, 256, 0, stream>>>(ew, feat, eidx, sum_wf, sum_w, sum_wn, E);

  int waves2  = (NS + 15) / 16;
  int blocks2 = (waves2 + 7) / 8;
  fsl_finalize_kernel<<<blocks2, 256, 0, stream>>>(sum_wf, sum_w, sum_wn, acc, cnt, NS);

  fsl_writeback_kernel<<<1, 32, 0, stream>>>(acc, cnt, (float*)d_out);
}